// LinearKernelAttention_82343112999021
// MI455X (gfx1250) — compile-verified
//
#include <hip/hip_runtime.h>
#include <stdint.h>

typedef float v2f __attribute__((ext_vector_type(2)));
typedef float v8f __attribute__((ext_vector_type(8)));

#define WAVES_PER_WG 8
#define B_ 2
#define H_ 16
#define S_ 2048
#define D_ 64
#define KBLK 16
#define NKB (S_ / KBLK) /* 128 */
#define ROWF 68  /* staged K/V row stride in floats: 64 data + 4 pad (16B) -> conflict-free b64 frag reads */
#define SSTR 18  /* per-wave S-tile staging stride: 8B-aligned b64 reads, conflict-free */

// Low 32 bits of a generic pointer to LDS = byte offset within LDS (aperture is in the high word).
__device__ __forceinline__ uint32_t lds_addr32(const void* p) {
    return (uint32_t)(uintptr_t)p;
}

// GVS-mode async copy: copies 16 bytes per lane from the global address saddr plus voffset to the LDS address in vdst, tracked by ASYNCcnt.
#define ASYNC_LOAD_B128(ldsoff, voff, base64)                         \
    asm volatile("global_load_async_to_lds_b128 %0, %1, %2"           \
                 :: "v"(ldsoff), "v"(voff), "s"(base64) : "memory")

// One wave32 computes a 16(query) x 64(D) output tile; the 8 waves of a WG share
// (b,h) and stream the same 16-key K/V tiles, staged cooperatively into LDS via
// async-to-LDS copies (double buffered, one tile ahead).
__global__ __launch_bounds__(WAVES_PER_WG * 32)
void lka_fused_norm_attn(const float* __restrict__ q,
                         const float* __restrict__ k,
                         const float* __restrict__ v,
                         const int*   __restrict__ mask,
                         float* __restrict__ out)
{
    __shared__ __align__(16) float kbuf[2][16 * ROWF];
    __shared__ __align__(16) float vbuf[2][16 * ROWF];
    __shared__ float sstage[WAVES_PER_WG][16 * SSTR];

    const int tid  = threadIdx.x;
    const int lane = tid & 31;
    const int wave = tid >> 5;
    const int n    = lane & 15;  // tile column / M-within-half
    const int g    = lane >> 4;  // half-wave selector

    // block -> (b, h, 128-query chunk); wave -> 16-query tile inside chunk
    const int qchunks = S_ / (WAVES_PER_WG * 16); // 16
    int blk = blockIdx.x;
    const int qc = blk % qchunks; blk /= qchunks;
    const int h  = blk % H_;
    const int b  = blk / H_;
    const int qbase = qc * (WAVES_PER_WG * 16) + wave * 16;

    const size_t slab = ((size_t)(b * H_ + h)) * S_ * D_;
    const float* Q = q + slab;
    const float* K = k + slab;
    const float* V = v + slab;
    float*       O = out + slab;
    const int*   M = mask + (size_t)b * S_ * S_;

    // ---- cooperative staging: a 16x64 f32 tile = 256 x 16B chunks; tid 0..255 covers all ----
    const int crow = tid >> 4;      // key row within tile
    const int ccol = tid & 15;      // 16B chunk within row
    uint32_t ldsKd[2], ldsVd[2];
    ldsKd[0] = lds_addr32(&kbuf[0][0]) + crow * (ROWF * 4) + ccol * 16;
    ldsKd[1] = lds_addr32(&kbuf[1][0]) + crow * (ROWF * 4) + ccol * 16;
    ldsVd[0] = lds_addr32(&vbuf[0][0]) + crow * (ROWF * 4) + ccol * 16;
    ldsVd[1] = lds_addr32(&vbuf[1][0]) + crow * (ROWF * 4) + ccol * 16;
    const uint64_t kbase64 = (uint64_t)(uintptr_t)K;
    const uint64_t vbase64 = (uint64_t)(uintptr_t)V;
    uint32_t vo = (uint32_t)(tid * 16);  // byte offset of this lane's chunk in the tile stream

    // prologue: tile 0 -> buffer 0
    ASYNC_LOAD_B128(ldsKd[0], vo, kbase64);
    ASYNC_LOAD_B128(ldsVd[0], vo, vbase64);
    vo += 4096;

    // ---- preload Q as A-fragments for V_WMMA_F32_16X16X4_F32 ----
    v2f aq[16];
#pragma unroll
    for (int f = 0; f < 16; ++f)
        aq[f] = *(const v2f*)(Q + (size_t)(qbase + n) * D_ + 4 * f + 2 * g);

    v8f acc[4];
#pragma unroll
    for (int t = 0; t < 4; ++t) acc[t] = v8f{};
    v8f p8 = v8f{};  // per-lane partial sum of squares (C layout)

    float* myS = sstage[wave];

#pragma unroll 2
    for (int kb = 0; kb < NKB; ++kb) {
        const int par = kb & 1;
        // own async writes (tile kb, issued last iteration) have landed
        asm volatile("s_wait_asynccnt 0x0" ::: "memory");
        // everyone's tile-kb writes landed; everyone done reading buf[1-par] (iter kb-1)
        __syncthreads();
        if (kb + 1 < NKB) {  // issue tile kb+1 into the buffer just freed
            ASYNC_LOAD_B128(ldsKd[par ^ 1], vo, kbase64);
            ASYNC_LOAD_B128(ldsVd[par ^ 1], vo, vbase64);
            vo += 4096;
        }

        const float* kp = &kbuf[par][0];
        const float* vp = &vbuf[par][0];
        const int kk = kb * KBLK;

        // ---- K tile B-fragments from LDS: lane holds K[kk+n][4f+2g+{0,1}] ----
        v2f bk[16];
#pragma unroll
        for (int f = 0; f < 16; ++f)
            bk[f] = *(const v2f*)(kp + n * ROWF + 4 * f + 2 * g);

        // ---- GEMM1: S(16x16) = Q · Kᵀ over D=64, 16 wmma steps ----
        v8f c = v8f{};
#pragma unroll
        for (int f = 0; f < 16; ++f)
            c = __builtin_amdgcn_wmma_f32_16x16x4_f32(false, aq[f], false, bk[f],
                                                      (short)0, c, false, false);

        // ---- epilogue: +1e-5, mask, sumsq, stage S to LDS (C layout: [r+8g][n]) ----
#pragma unroll
        for (int r = 0; r < 8; ++r) {
            const int qrow = qbase + r + 8 * g;
            const int mv = M[(size_t)qrow * S_ + kk + n];
            const float s = (mv == 0) ? 0.0f : (c[r] + 1e-5f);
            p8[r] += s * s;
            myS[(r + 8 * g) * SSTR + n] = s;
        }
        asm volatile("s_wait_dscnt 0" ::: "memory");

        // ---- reload S as A-fragments: lane holds S[n][4f2+2g+{0,1}] ----
        v2f sa[4];
#pragma unroll
        for (int f2 = 0; f2 < 4; ++f2)
            sa[f2] = *(const v2f*)(myS + n * SSTR + 4 * f2 + 2 * g);
        asm volatile("" ::: "memory");

        // ---- GEMM2: acc(16x64) += S · V; V frags from LDS ----
#pragma unroll
        for (int t = 0; t < 4; ++t) {
            v2f bv[4];
#pragma unroll
            for (int f2 = 0; f2 < 4; ++f2) {
                bv[f2][0] = vp[(4 * f2 + 2 * g + 0) * ROWF + 16 * t + n];
                bv[f2][1] = vp[(4 * f2 + 2 * g + 1) * ROWF + 16 * t + n];
            }
#pragma unroll
            for (int f2 = 0; f2 < 4; ++f2)
                acc[t] = __builtin_amdgcn_wmma_f32_16x16x4_f32(false, sa[f2], false, bv[f2],
                                                               (short)0, acc[t], false, false);
        }
    }

    // ---- rowwise L2 norm: butterfly within each 16-lane half lands the
    //      per-row sum on exactly the lanes/regs holding that row's outputs ----
#pragma unroll
    for (int off = 1; off < 16; off <<= 1) {
#pragma unroll
        for (int r = 0; r < 8; ++r)
            p8[r] += __shfl_xor(p8[r], off, 32);
    }
    float sc[8];
#pragma unroll
    for (int r = 0; r < 8; ++r)
        sc[r] = 1.0f / fmaxf(sqrtf(p8[r]), 1e-12f);

    // ---- scaled writeout: O[qbase + r + 8g][16t + n] ----
#pragma unroll
    for (int t = 0; t < 4; ++t)
#pragma unroll
        for (int r = 0; r < 8; ++r)
            O[(size_t)(qbase + r + 8 * g) * D_ + 16 * t + n] = acc[t][r] * sc[r];
}

extern "C" void kernel_launch(void* const* d_in, const int* in_sizes, int n_in,
                              void* d_out, int out_size, void* d_ws, size_t ws_size,
                              hipStream_t stream) {
    (void)in_sizes; (void)n_in; (void)out_size; (void)d_ws; (void)ws_size;
    const float* q    = (const float*)d_in[0];
    const float* k    = (const float*)d_in[1];
    const float* v    = (const float*)d_in[2];
    const int*   mask = (const int*)d_in[3];
    float*       out  = (float*)d_out;

    dim3 grid(B_ * H_ * (S_ / (WAVES_PER_WG * 16))); // 2*16*16 = 512
    dim3 block(WAVES_PER_WG * 32);                   // 256 threads = 8 waves
    lka_fused_norm_attn<<<grid, block, 0, stream>>>(q, k, v, mask, out);
}